// SAGEConvOp_61976378081387
// MI455X (gfx1250) — compile-verified
//
#include <hip/hip_runtime.h>

#define N_NODES 100000
#define N_EDGES 1600000
#define FEAT    128
#define LAYERS  4

typedef __attribute__((ext_vector_type(2))) float v2f;
typedef __attribute__((ext_vector_type(4))) float v4f;
typedef __attribute__((ext_vector_type(8))) float v8f;

#define ROWS_PER_BLOCK 64
#define LDS_STRIDE     132   // 128 + 4 pad: A-frag reads hit distinct LDS banks (row = 528B, 16B aligned)
#define GEMM_THREADS   128   // 4 waves, 16 rows each

#if defined(__gfx1250__) && __has_builtin(__builtin_amdgcn_global_load_async_to_lds_b128) && __has_builtin(__builtin_amdgcn_s_wait_asynccnt)
#define USE_ASYNC_LDS 1
#else
#define USE_ASYNC_LDS 0
#endif

#if USE_ASYNC_LDS
typedef int vi4 __attribute__((vector_size(16)));                 // <4 x i32>
typedef __attribute__((address_space(1))) vi4 g_vi4;              // global
typedef __attribute__((address_space(3))) vi4 l_vi4;              // LDS
#endif

// out[m,:] = relu( x[m,:] @ W1 + (x2 ? x2[m,:] @ W2 : 0) + bias ), W row-major [K=128][N=128]
__global__ __launch_bounds__(GEMM_THREADS) void sage_gemm(
    const float* __restrict__ x,  const float* __restrict__ W1,
    const float* __restrict__ x2, const float* __restrict__ W2,
    const float* __restrict__ bias, float* __restrict__ out, int nrows)
{
    __shared__ float lds[ROWS_PER_BLOCK * LDS_STRIDE];

    const int tid  = threadIdx.x;
    const int wave = tid >> 5;
    const int lane = tid & 31;
    const int half = lane >> 4;   // 0: lanes 0-15, 1: lanes 16-31
    const int l16  = lane & 15;
    const int m0   = blockIdx.x * ROWS_PER_BLOCK;  // block row base
    const int wm   = wave * 16;                    // wave row offset in block
    const bool full_tile = (m0 + ROWS_PER_BLOCK) <= nrows;

    v8f acc[8] = {};  // 8 16x16 output tiles covering all 128 columns

    for (int pass = 0; pass < 2; ++pass) {
        const float* xp = (pass == 0) ? x  : x2;
        const float* Wm = (pass == 0) ? W1 : W2;
        if (xp == nullptr) break;   // uniform branch (kernel arg)

        __syncthreads();  // protect LDS from previous pass readers
#if USE_ASYNC_LDS
        if (full_tile) {
            // Async DMA: global -> LDS, bypassing VGPRs (ASYNCcnt-tracked)
            for (int i = tid; i < ROWS_PER_BLOCK * (FEAT / 4); i += GEMM_THREADS) {
                int r  = i >> 5;            // row in tile
                int c4 = (i & 31) << 2;     // float4 column
                const float* g = xp + (size_t)(m0 + r) * FEAT + c4;
                float*       l = lds + r * LDS_STRIDE + c4;
                __builtin_amdgcn_global_load_async_to_lds_b128(
                    (g_vi4*)g, (l_vi4*)l, 0, 0);
            }
            __builtin_amdgcn_s_wait_asynccnt(0);
        } else
#endif
        {
            // Guarded manual path (ragged tail block)
            for (int i = tid; i < ROWS_PER_BLOCK * (FEAT / 4); i += GEMM_THREADS) {
                int r  = i >> 5;
                int c4 = (i & 31) << 2;
                int gr = m0 + r;
                v4f v = {};
                if (gr < nrows) v = *(const v4f*)(xp + (size_t)gr * FEAT + c4);
                float* dstl = lds + r * LDS_STRIDE + c4;
                dstl[0] = v.x; dstl[1] = v.y; dstl[2] = v.z; dstl[3] = v.w;
            }
        }
        __syncthreads();

        const float* arow = lds + (wm + l16) * LDS_STRIDE;
        #pragma unroll 2
        for (int k0 = 0; k0 < FEAT; k0 += 4) {
            // A frag (16x4): VGPR0 = K=k0 / k0+2 per lane half, VGPR1 = K=k0+1 / k0+3
            v2f a;
            a.x = arow[k0 + 2 * half + 0];
            a.y = arow[k0 + 2 * half + 1];
            #pragma unroll
            for (int c = 0; c < 8; ++c) {
                // B frag (4x16): rows k0..k0+3, cols c*16..c*16+15
                const float* wp = Wm + (size_t)(k0 + 2 * half) * FEAT + c * 16 + l16;
                v2f b;
                b.x = wp[0];
                b.y = wp[FEAT];
                acc[c] = __builtin_amdgcn_wmma_f32_16x16x4_f32(
                    false, a, false, b, (short)0, acc[c], false, false);
            }
        }
    }

    // Epilogue: bias + relu + store.  C/D layout: lane = col (mod 16),
    // VGPR i -> row i (lanes 0-15) or row 8+i (lanes 16-31).
    #pragma unroll
    for (int c = 0; c < 8; ++c) {
        float bv = bias[c * 16 + l16];
        #pragma unroll
        for (int i = 0; i < 8; ++i) {
            int gr = m0 + wm + i + 8 * half;
            if (gr < nrows) {
                float v = acc[c][i] + bv;
                out[(size_t)gr * FEAT + c * 16 + l16] = v > 0.0f ? v : 0.0f;
            }
        }
    }
}

// pooled = 0  (bit pattern 0 == 0.0f, valid identity since h_pool >= 0)
__global__ __launch_bounds__(256) void sage_zero(float* __restrict__ p, int n4)
{
    int i = blockIdx.x * 256 + threadIdx.x;
    if (i < n4) { v4f z = {}; *(v4f*)(p + (size_t)i * 4) = z; }
}

// segment-max via uint atomicMax (values are relu outputs >= 0, bits order like floats)
__global__ __launch_bounds__(256) void sage_scatter_max(
    const float* __restrict__ h, const int* __restrict__ src,
    const int* __restrict__ dst, unsigned int* __restrict__ pooled_bits)
{
    long long idx = (long long)blockIdx.x * 256 + threadIdx.x;
    long long e = idx >> 5;
    if (e >= N_EDGES) return;
    int chunk = (int)(idx & 31) << 2;   // 4 floats per thread
    int s = src[e];
    int d = dst[e];
    v4f v = *(const v4f*)(h + (size_t)s * FEAT + chunk);
    unsigned int* p = pooled_bits + (size_t)d * FEAT + chunk;
    atomicMax(p + 0, __float_as_uint(v.x));
    atomicMax(p + 1, __float_as_uint(v.y));
    atomicMax(p + 2, __float_as_uint(v.z));
    atomicMax(p + 3, __float_as_uint(v.w));
}

extern "C" void kernel_launch(void* const* d_in, const int* in_sizes, int n_in,
                              void* d_out, int out_size, void* d_ws, size_t ws_size,
                              hipStream_t stream) {
    const float* feats = (const float*)d_in[0];
    const float* Wp    = (const float*)d_in[1];   // [L,128,128]
    const float* bp    = (const float*)d_in[2];   // [L,128]
    const float* Ws    = (const float*)d_in[3];
    const float* Wn    = (const float*)d_in[4];
    const float* b     = (const float*)d_in[5];
    const int*   src   = (const int*)d_in[6];
    const int*   dst   = (const int*)d_in[7];

    const size_t NF = (size_t)N_NODES * FEAT;
    float* xb0    = (float*)d_ws;
    float* xb1    = xb0 + NF;
    float* hpool  = xb1 + NF;
    float* pooled = hpool + NF;

    const int gemm_blocks    = (N_NODES + ROWS_PER_BLOCK - 1) / ROWS_PER_BLOCK;
    const int zero_blocks    = (int)((NF / 4 + 255) / 256);
    const long long sc_thr   = (long long)N_EDGES * 32;
    const int scatter_blocks = (int)((sc_thr + 255) / 256);

    const float* xin = feats;
    float* xping[2] = { xb0, xb1 };
    for (int l = 0; l < LAYERS; ++l) {
        const float* Wp_l = Wp + (size_t)l * FEAT * FEAT;
        const float* bp_l = bp + (size_t)l * FEAT;
        const float* Ws_l = Ws + (size_t)l * FEAT * FEAT;
        const float* Wn_l = Wn + (size_t)l * FEAT * FEAT;
        const float* b_l  = b  + (size_t)l * FEAT;

        // h_pool = relu(x @ Wp + bp)
        sage_gemm<<<gemm_blocks, GEMM_THREADS, 0, stream>>>(
            xin, Wp_l, nullptr, nullptr, bp_l, hpool, N_NODES);

        // pooled = segment_max(h_pool[src], dst), empty -> 0
        sage_zero<<<zero_blocks, 256, 0, stream>>>(pooled, (int)(NF / 4));
        sage_scatter_max<<<scatter_blocks, 256, 0, stream>>>(
            hpool, src, dst, (unsigned int*)pooled);

        // x = relu(x @ Ws + pooled @ Wn + b)
        float* o = (l == LAYERS - 1) ? (float*)d_out : xping[l & 1];
        sage_gemm<<<gemm_blocks, GEMM_THREADS, 0, stream>>>(
            xin, Ws_l, pooled, Wn_l, b_l, o, N_NODES);
        xin = o;
    }
}